// Dropblock_48387101557183
// MI455X (gfx1250) — compile-verified
//
#include <hip/hip_runtime.h>
#include <stdint.h>

// Native vector types (HIP's float4 is a struct; nontemporal builtins need
// scalar/vector types).
typedef float        f4    __attribute__((ext_vector_type(4)));
typedef unsigned int u32x4 __attribute__((ext_vector_type(4)));
typedef int          i32x4 __attribute__((ext_vector_type(4)));
typedef int          i32x8 __attribute__((ext_vector_type(8)));

#define BS     7
#define DROPP  0.1f
#define UH     58
#define UW     58
#define UELEMS (UH * UW)   // 3364 floats = 13456 bytes
#define IMG_H  64
#define IMG_W  64
#define HW     4096

// ---------------------------------------------------------------------------
// Stage 1: build the premultiplied keep-mask m[64*64] = keep(y,x) * 4096/sum.
// Single workgroup. u (58x58 f32) is DMA'd into LDS with the Tensor Data
// Mover (tensor_load_to_lds, tracked by TENSORcnt), then each thread tests
// its 16 pixels' 7x7 coverage windows out of LDS and the block tree-reduces
// the keep count.
// ---------------------------------------------------------------------------
__global__ __launch_bounds__(256) void dropblock_mask_kernel(
    const float* __restrict__ u, float* __restrict__ m)
{
    __shared__ float smem[UELEMS + 256];
    float* ulds = smem;            // LDS offset 0 — matches D#.lds_addr = 0
    float* red  = smem + UELEMS;
    const int tid = threadIdx.x;

#if defined(__gfx1250__) && __has_builtin(__builtin_amdgcn_tensor_load_to_lds)
    if (tid < 32) {  // one wave issues the TDM op (EXEC-independent DMA)
        const uint64_t ga = (uint64_t)(uintptr_t)u;
        // D# group 0 (128b): count=1 | lds_addr=0 | global_addr | type=2
        u32x4 g0;
        g0[0] = 1u;                                   // count=1, user desc
        g0[1] = 0u;                                   // lds_addr = 0 bytes
        g0[2] = (unsigned)(ga & 0xFFFFFFFFu);         // global_addr[31:0]
        g0[3] = (unsigned)((ga >> 32) & 0x1FFFFFFu)   // global_addr[56:32]
              | (2u << 30);                           // type = 2 ("image")
        // D# group 1 (256b): data_size=4B, tensor 58x58, tile 58x58
        i32x8 g1;
        g1[0] = (int)(2u << 16);                      // data_size=2 -> 4 bytes
        g1[1] = (int)((unsigned)UW << 16);            // tensor_dim0 = 58
        g1[2] = (int)((unsigned)UH << 16);            // tensor_dim1 = 58
        g1[3] = (int)((unsigned)UW << 16);            // tile_dim0   = 58
        g1[4] = (int)UW;                              // tile_dim1   = 58
        g1[5] = (int)UW;                              // tensor_dim0_stride=58
        g1[6] = (int)((unsigned)(UW * UH) << 16);     // tensor_dim1_stride lo
        g1[7] = 0;
        i32x4 gz4 = {0, 0, 0, 0};                     // groups 2/3 unused (2D)
        i32x8 gz8 = {0, 0, 0, 0, 0, 0, 0, 0};         // extra group (clang-23 form)
        __builtin_amdgcn_tensor_load_to_lds(g0, g1, gz4, gz4, gz8, 0);
        __builtin_amdgcn_s_wait_tensorcnt(0);         // TENSORcnt == 0
    }
    __syncthreads();
#else
    for (int i = tid; i < UELEMS; i += 256) ulds[i] = u[i];
    __syncthreads();
#endif

    float keep[HW / 256];
    float lsum = 0.0f;
#pragma unroll
    for (int k = 0; k < HW / 256; ++k) {
        const int p = tid + (k << 8);
        const int y = p >> 6;
        const int x = p & 63;
        int i0 = y - (BS - 1); if (i0 < 0) i0 = 0;
        int i1 = y;            if (i1 > UH - 1) i1 = UH - 1;
        int j0 = x - (BS - 1); if (j0 < 0) j0 = 0;
        int j1 = x;            if (j1 > UW - 1) j1 = UW - 1;
        float kp = 1.0f;
        for (int i = i0; i <= i1; ++i)
            for (int j = j0; j <= j1; ++j)
                if (ulds[i * UW + j] < DROPP) kp = 0.0f;
        keep[k] = kp;
        lsum += kp;
    }

    red[tid] = lsum;
    __syncthreads();
    for (int s = 128; s > 0; s >>= 1) {
        if (tid < s) red[tid] += red[tid + s];
        __syncthreads();
    }
    const float scale = (float)HW / red[0];

#pragma unroll
    for (int k = 0; k < HW / 256; ++k) {
        const int p = tid + (k << 8);
        m[p] = keep[k] * scale;
    }
}

// ---------------------------------------------------------------------------
// Stage 2: out = x * m[y,x], 64M f32 elements, pure HBM streaming.
// float4 (B128) loads/stores; nontemporal hints on the 256MB-each-way stream
// (working set > 192MB L2, touched once), regular-temporal load for the 16KB
// mask (re-read by every block -> cache hit).
// ---------------------------------------------------------------------------
__global__ __launch_bounds__(256) void dropblock_apply_kernel(
    const f4* __restrict__ x, const float* __restrict__ m,
    f4* __restrict__ out, int total4)
{
    const f4* m4 = (const f4*)m;   // 1024 float4 per (n,c) plane
    int i = blockIdx.x * 256 + threadIdx.x;
    const int stride = gridDim.x * 256;
    for (; i < total4; i += stride) {
        f4 xv = __builtin_nontemporal_load(&x[i]);
        f4 mv = m4[i & 1023];      // (plane offset / 4) since HW = 4096
        f4 r  = xv * mv;
        __builtin_nontemporal_store(r, &out[i]);
    }
}

extern "C" void kernel_launch(void* const* d_in, const int* in_sizes, int n_in,
                              void* d_out, int out_size, void* d_ws, size_t ws_size,
                              hipStream_t stream) {
    const float* x = (const float*)d_in[0];   // [64,256,64,64] f32
    const float* u = (const float*)d_in[1];   // [58,58] f32
    float* out = (float*)d_out;
    float* m   = (float*)d_ws;                // 4096 floats of scratch

    dropblock_mask_kernel<<<1, 256, 0, stream>>>(u, m);

    const int total4 = out_size / 4;          // 16,777,216 float4
    int grid = 16384;                          // 4 float4 iters per thread
    dropblock_apply_kernel<<<grid, 256, 0, stream>>>(
        (const f4*)x, m, (f4*)out, total4);
}